// PharmMatchNetFast_49950469653202
// MI455X (gfx1250) — compile-verified
//
#include <hip/hip_runtime.h>
#include <math.h>

// ---------------------------------------------------------------------------
// PharmMatchNet forward, fused for gfx1250 (MI455X).
// All GEMMs run on V_WMMA_F32_16X16X4_F32 (full fp32, matches reference).
// ---------------------------------------------------------------------------

typedef __attribute__((ext_vector_type(2))) float v2f;
typedef __attribute__((ext_vector_type(8))) float v8f;

#define D_     128
#define B_     128
#define L_     128
#define P_     1024
#define SCALE_ 0.08838834764831845f   // 1/sqrt(128)
#define NEGF   (-3.402823466e38f)

__device__ __forceinline__ v8f wmma4(v8f c, v2f a, v2f b) {
  // D = A(16x4,f32) * B(4x16,f32) + C(16x16,f32)
  return __builtin_amdgcn_wmma_f32_16x16x4_f32(
      /*neg_a=*/false, a, /*neg_b=*/false, b,
      /*c_mod=*/(short)0, c, /*reuse_a=*/false, /*reuse_b=*/false);
}

// C(16x16) += A(16xK, row-major, stride lda) * B(KxN tile, row-major, stride ldb)
// Wave-level: lane holds (m = lane&15, n = lane&15) with K split per half-wave.
__device__ __forceinline__ v8f gemm_tile(const float* A, int lda,
                                         const float* Bg, int ldb,
                                         int K, int lane, v8f c) {
  const int m = lane & 15;
  const int koff = (lane >> 4) << 1;  // 0 for lanes 0-15, 2 for lanes 16-31
  for (int k = 0; k < K; k += 4) {
    const int kk = k + koff;
    v2f a, b;
    a.x = A[m * lda + kk];
    a.y = A[m * lda + kk + 1];
    b.x = Bg[kk * ldb + m];        // column index n == m (same lane pattern)
    b.y = Bg[(kk + 1) * ldb + m];
    c = wmma4(c, a, b);
  }
  return c;
}

// Same but B given transposed: Bt is (N x K) row-major (e.g. hP used as B^T).
__device__ __forceinline__ v8f gemm_tile_bt(const float* A, int lda,
                                            const float* Bt, int ldbt,
                                            int K, int lane, v8f c) {
  const int m = lane & 15;
  const int koff = (lane >> 4) << 1;
  for (int k = 0; k < K; k += 4) {
    const int kk = k + koff;
    v2f a, b;
    a.x = A[m * lda + kk];
    a.y = A[m * lda + kk + 1];
    b.x = Bt[m * ldbt + kk];
    b.y = Bt[m * ldbt + kk + 1];
    c = wmma4(c, a, b);
  }
  return c;
}

__device__ __forceinline__ float gelu_exact(float x) {
  return 0.5f * x * (1.0f + erff(x * 0.70710678118654752440f));
}

// ---------------------------------------------------------------------------
// Ligand encoder: hL = LN(gelu(x@w1+b1)@w2+b2 + emb) @ projw
// One block = 16 token rows, 8 waves = 8 column tiles of D=128.
// ---------------------------------------------------------------------------
__global__ __launch_bounds__(256) void encode_l_kernel(
    const float* __restrict__ l_x, const int* __restrict__ l_typ,
    const float* __restrict__ lt_emb,
    const float* __restrict__ w1, const float* __restrict__ b1,
    const float* __restrict__ w2, const float* __restrict__ b2,
    const float* __restrict__ ln_g, const float* __restrict__ ln_b,
    const float* __restrict__ projw,
    float* __restrict__ hL) {
  __shared__ float Xs[16][8];
  __shared__ float Ts[16][132];
  __shared__ float T2[16][132];
  __shared__ int typs[16];
  __shared__ float mu_s[16], rs_s[16];

  const int tid = threadIdx.x;
  const int lane = tid & 31;
  const int wave = tid >> 5;
  const int row0 = blockIdx.x * 16;
  const int n0 = wave * 16;
  const int n = lane & 15;
  const int mb = (lane >> 4) * 8;

  if (tid < 16 * 8) Xs[tid >> 3][tid & 7] = l_x[row0 * 8 + tid];
  if (tid < 16) {
    int t = l_typ[row0 + tid];
    typs[tid] = t < 0 ? 0 : (t > 255 ? 255 : t);
  }
  __syncthreads();

  {  // T1 = gelu(X @ w1 + b1)
    v8f c = {};
    c = gemm_tile(&Xs[0][0], 8, w1 + n0, D_, 8, lane, c);
    const float bb = b1[n0 + n];
#pragma unroll
    for (int r = 0; r < 8; ++r) Ts[mb + r][n0 + n] = gelu_exact(c[r] + bb);
  }
  __syncthreads();

  {  // T2 = T1 @ w2 + b2 + lt_emb[typ]
    v8f c = {};
    c = gemm_tile(&Ts[0][0], 132, w2 + n0, D_, D_, lane, c);
    const float bb = b2[n0 + n];
#pragma unroll
    for (int r = 0; r < 8; ++r)
      T2[mb + r][n0 + n] = c[r] + bb + lt_emb[typs[mb + r] * D_ + n0 + n];
  }
  __syncthreads();

  if (tid < 16) {  // row LayerNorm stats
    float s = 0.f;
    for (int j = 0; j < D_; ++j) s += T2[tid][j];
    const float mu = s * 0.0078125f;
    float v = 0.f;
    for (int j = 0; j < D_; ++j) { float d = T2[tid][j] - mu; v += d * d; }
    mu_s[tid] = mu;
    rs_s[tid] = rsqrtf(v * 0.0078125f + 1e-5f);
  }
  __syncthreads();
  for (int idx = tid; idx < 16 * D_; idx += 256) {
    const int r = idx >> 7, cc = idx & 127;
    Ts[r][cc] = (T2[r][cc] - mu_s[r]) * rs_s[r] * ln_g[cc] + ln_b[cc];
  }
  __syncthreads();

  {  // hL tile = LN(...) @ projw
    v8f c = {};
    c = gemm_tile(&Ts[0][0], 132, projw + n0, D_, D_, lane, c);
#pragma unroll
    for (int r = 0; r < 8; ++r)
      hL[((size_t)(row0 + mb + r)) * D_ + n0 + n] = c[r];
  }
}

// ---------------------------------------------------------------------------
// Pocket encoder (+ fused sigma head).
// ---------------------------------------------------------------------------
__global__ __launch_bounds__(256) void encode_p_kernel(
    const float* __restrict__ p_x, const int* __restrict__ p_typ,
    const float* __restrict__ p_score, const float* __restrict__ p_rad,
    const float* __restrict__ pt_emb,
    const float* __restrict__ w1, const float* __restrict__ b1,
    const float* __restrict__ w2, const float* __restrict__ b2,
    const float* __restrict__ psw1, const float* __restrict__ psb1,
    const float* __restrict__ psw2, const float* __restrict__ psb2,
    const float* __restrict__ ln_g, const float* __restrict__ ln_b,
    const float* __restrict__ projw,
    const float* __restrict__ sgw1, const float* __restrict__ sgb1,
    const float* __restrict__ sgw2, const float* __restrict__ sgb2,
    float* __restrict__ hP, float* __restrict__ sig_out) {
  __shared__ float Xs[16 * 192];  // also reused as 16x132 hP tile later
  __shared__ float Ts[16][132];
  __shared__ float T2[16][132];
  __shared__ float Tps[16][132];
  __shared__ int typs[16];
  __shared__ float scr[16];
  __shared__ float mu_s[16], rs_s[16];

  const int tid = threadIdx.x;
  const int lane = tid & 31;
  const int wave = tid >> 5;
  const int row0 = blockIdx.x * 16;
  const int n0 = wave * 16;
  const int n = lane & 15;
  const int mb = (lane >> 4) * 8;

  for (int idx = tid; idx < 16 * 192; idx += 256)
    Xs[idx] = p_x[(size_t)row0 * 192 + idx];
  if (tid < 16) {
    int t = p_typ[row0 + tid];
    typs[tid] = t < 0 ? 0 : (t > 255 ? 255 : t);
    scr[tid] = p_score[row0 + tid];
  }
  __syncthreads();

  {  // T1 = gelu(X @ w1 + b1), K=192
    v8f c = {};
    c = gemm_tile(Xs, 192, w1 + n0, D_, 192, lane, c);
    const float bb = b1[n0 + n];
#pragma unroll
    for (int r = 0; r < 8; ++r) Ts[mb + r][n0 + n] = gelu_exact(c[r] + bb);
  }
  // Tps = gelu(score * psw1 + psb1)  (rank-1 expansion, elementwise)
  for (int idx = tid; idx < 16 * D_; idx += 256) {
    const int r = idx >> 7, cc = idx & 127;
    Tps[r][cc] = gelu_exact(scr[r] * psw1[cc] + psb1[cc]);
  }
  __syncthreads();

  {  // T2 = T1@w2 + Tps@psw2 + b2 + psb2 + pt_emb[typ]
    v8f c = {};
    c = gemm_tile(&Ts[0][0], 132, w2 + n0, D_, D_, lane, c);
    c = gemm_tile(&Tps[0][0], 132, psw2 + n0, D_, D_, lane, c);
    const float bb = b2[n0 + n] + psb2[n0 + n];
#pragma unroll
    for (int r = 0; r < 8; ++r)
      T2[mb + r][n0 + n] = c[r] + bb + pt_emb[typs[mb + r] * D_ + n0 + n];
  }
  __syncthreads();

  if (tid < 16) {  // LayerNorm stats
    float s = 0.f;
    for (int j = 0; j < D_; ++j) s += T2[tid][j];
    const float mu = s * 0.0078125f;
    float v = 0.f;
    for (int j = 0; j < D_; ++j) { float d = T2[tid][j] - mu; v += d * d; }
    mu_s[tid] = mu;
    rs_s[tid] = rsqrtf(v * 0.0078125f + 1e-5f);
  }
  __syncthreads();
  for (int idx = tid; idx < 16 * D_; idx += 256) {
    const int r = idx >> 7, cc = idx & 127;
    Ts[r][cc] = (T2[r][cc] - mu_s[r]) * rs_s[r] * ln_g[cc] + ln_b[cc];
  }
  __syncthreads();

  float* Hs = Xs;  // reuse (16 x 132 view); Xs no longer needed
  {                // hP tile = LN(...) @ projw
    v8f c = {};
    c = gemm_tile(&Ts[0][0], 132, projw + n0, D_, D_, lane, c);
#pragma unroll
    for (int r = 0; r < 8; ++r) {
      const float v = c[r];
      hP[((size_t)(row0 + mb + r)) * D_ + n0 + n] = v;
      Hs[(mb + r) * 132 + n0 + n] = v;
    }
  }
  __syncthreads();

  {  // sigma head: Tg = gelu(hP_tile @ sgw1 + sgb1)
    v8f c = {};
    c = gemm_tile(Hs, 132, sgw1 + n0, D_, D_, lane, c);
    const float bb = sgb1[n0 + n];
#pragma unroll
    for (int r = 0; r < 8; ++r) Tps[mb + r][n0 + n] = gelu_exact(c[r] + bb);
  }
  __syncthreads();

  if (tid < 16) {  // sigma = softplus(Tg . sgw2 + sgb2) + 1e-3 + max(rad,0)
    float s = 0.f;
    for (int j = 0; j < D_; ++j) s += Tps[tid][j] * sgw2[j];
    s += sgb2[0];
    const float sp = fmaxf(s, 0.f) + log1pf(expf(-fabsf(s)));
    const float rad = p_rad[row0 + tid];
    sig_out[row0 + tid] = sp + 1e-3f + fmaxf(rad, 0.f);
  }
}

// ---------------------------------------------------------------------------
// Logits + dustbin + masked log-softmax + sharp/entropy partials.
// One block per (batch, 16 ligand rows): 128*8 = 1024 blocks.
// ---------------------------------------------------------------------------
__global__ __launch_bounds__(256) void logits_kernel(
    const float* __restrict__ hL, const float* __restrict__ hP,
    const float* __restrict__ dustbin,
    const unsigned char* __restrict__ l_mask,
    const unsigned char* __restrict__ p_mask,
    float* __restrict__ logW, float* __restrict__ partials) {
  __shared__ float As[16][132];
  __shared__ float Lg[16][1032];  // 1025 used
  __shared__ float pms[P_];
  __shared__ float lms[16];
  __shared__ float rowSharp[16], rowEnt[16];

  const int tid = threadIdx.x;
  const int lane = tid & 31;
  const int wave = tid >> 5;
  const int b = blockIdx.x >> 3;
  const int l0 = (blockIdx.x & 7) * 16;
  const int n = lane & 15;
  const int mb = (lane >> 4) * 8;

  for (int idx = tid; idx < 16 * D_; idx += 256)
    As[idx >> 7][idx & 127] =
        hL[((size_t)(b * L_ + l0 + (idx >> 7))) * D_ + (idx & 127)];
  for (int idx = tid; idx < P_; idx += 256)
    pms[idx] = p_mask[b * P_ + idx] ? 1.0f : 0.0f;
  if (tid < 16) lms[tid] = l_mask[b * L_ + l0 + tid] ? 1.0f : 0.0f;
  __syncthreads();

  const float* hPb = hP + (size_t)b * P_ * D_;
  for (int pt = wave; pt < P_ / 16; pt += 8) {  // uniform per wave
    const int p0 = pt * 16;
    v8f c = {};
    c = gemm_tile_bt(&As[0][0], 132, hPb + (size_t)p0 * D_, D_, D_, lane, c);
    const float pm = pms[p0 + n];
#pragma unroll
    for (int r = 0; r < 8; ++r) {
      const float keep = pm * lms[mb + r];
      Lg[mb + r][p0 + n] = (keep != 0.0f) ? c[r] * SCALE_ : NEGF;
    }
  }
  if (tid < 16) {  // dustbin column (masked by l_mask only)
    float d = 0.f;
    for (int k = 0; k < D_; ++k) d += As[tid][k] * dustbin[k];
    Lg[tid][P_] = (lms[tid] != 0.f) ? d * SCALE_ : NEGF;
  }
  __syncthreads();

  // log-softmax over 1025 cols; each wave handles 2 rows.
  for (int rr = 0; rr < 2; ++rr) {
    const int row = wave * 2 + rr;
    float mx = NEGF;
    for (int j = lane; j < P_ + 1; j += 32) mx = fmaxf(mx, Lg[row][j]);
    for (int o = 16; o; o >>= 1) mx = fmaxf(mx, __shfl_xor(mx, o, 32));
    float se = 0.f;
    for (int j = lane; j < P_ + 1; j += 32) se += expf(Lg[row][j] - mx);
    for (int o = 16; o; o >>= 1) se += __shfl_xor(se, o, 32);
    const float lse = logf(se);
    float mxw = 0.f, ent = 0.f;
    float* orow = logW + ((size_t)(b * L_ + l0 + row)) * (P_ + 1);
    for (int j = lane; j < P_ + 1; j += 32) {
      const float lw = Lg[row][j] - mx - lse;
      orow[j] = lw;
      const float w = expf(lw);
      const float wm = (j < P_) ? w * pms[j] : w;
      mxw = fmaxf(mxw, wm);
      ent -= wm * lw;
    }
    for (int o = 16; o; o >>= 1) {
      mxw = fmaxf(mxw, __shfl_xor(mxw, o, 32));
      ent += __shfl_xor(ent, o, 32);
    }
    if (lane == 0) {
      rowSharp[row] = mxw * lms[row];
      rowEnt[row] = ent * lms[row];
    }
  }
  __syncthreads();
  if (tid == 0) {  // deterministic per-block partial (no atomics)
    float s = 0.f, e = 0.f;
    for (int r = 0; r < 16; ++r) { s += rowSharp[r]; e += rowEnt[r]; }
    partials[blockIdx.x * 2 + 0] = s;
    partials[blockIdx.x * 2 + 1] = e;
  }
}

// ---------------------------------------------------------------------------
// Pooled embeddings + retrieval projection + l2norm.
// Blocks 0..127 -> ligand, 128..255 -> pocket. 128 threads = one per channel.
// ---------------------------------------------------------------------------
__global__ __launch_bounds__(128) void pooled_kernel(
    const float* __restrict__ hL, const float* __restrict__ hP,
    const unsigned char* __restrict__ l_mask,
    const unsigned char* __restrict__ p_mask,
    const float* __restrict__ retr_l, const float* __restrict__ retr_p,
    float* __restrict__ lz, float* __restrict__ pz) {
  __shared__ float g[D_];
  __shared__ float part[4];
  const int tid = threadIdx.x;
  const bool lig = blockIdx.x < B_;
  const int b = lig ? blockIdx.x : blockIdx.x - B_;
  const float* h = lig ? hL + (size_t)b * L_ * D_ : hP + (size_t)b * P_ * D_;
  const unsigned char* msk = lig ? l_mask + b * L_ : p_mask + b * P_;
  const int N = lig ? L_ : P_;

  float acc = 0.f, cnt = 0.f;
  for (int i = 0; i < N; ++i) {
    const float m = msk[i] ? 1.f : 0.f;
    acc += h[(size_t)i * D_ + tid] * m;
    cnt += m;
  }
  g[tid] = acc / fmaxf(cnt, 1.f);
  __syncthreads();

  const float* rw = lig ? retr_l : retr_p;
  float z = 0.f;
  for (int k = 0; k < D_; ++k) z += g[k] * rw[k * D_ + tid];

  float ss = z * z;
  for (int o = 16; o; o >>= 1) ss += __shfl_xor(ss, o, 32);
  if ((tid & 31) == 0) part[tid >> 5] = ss;
  __syncthreads();
  const float tot = part[0] + part[1] + part[2] + part[3];
  const float inv = 1.f / fmaxf(sqrtf(tot), 1e-12f);
  (lig ? lz : pz)[b * D_ + tid] = z * inv;
}

// ---------------------------------------------------------------------------
// Finalize sharp / neg-entropy per batch (deterministic partial reduction).
// ---------------------------------------------------------------------------
__global__ void finalize_kernel(const unsigned char* __restrict__ l_mask,
                                const float* __restrict__ partials,
                                float* __restrict__ sharp,
                                float* __restrict__ negent) {
  const int b = threadIdx.x;
  if (b >= B_) return;
  float cnt = 0.f;
  for (int l = 0; l < L_; ++l) cnt += l_mask[b * L_ + l] ? 1.f : 0.f;
  const float denom = fmaxf(cnt, 1.f);
  float s = 0.f, e = 0.f;
  for (int t = 0; t < 8; ++t) {
    s += partials[(b * 8 + t) * 2 + 0];
    e += partials[(b * 8 + t) * 2 + 1];
  }
  sharp[b] = s / denom;
  negent[b] = -(e / denom);
}

// ---------------------------------------------------------------------------
extern "C" void kernel_launch(void* const* d_in, const int* in_sizes, int n_in,
                              void* d_out, int out_size, void* d_ws,
                              size_t ws_size, hipStream_t stream) {
  (void)in_sizes; (void)n_in; (void)out_size; (void)ws_size;

  const float* l_x     = (const float*)d_in[0];
  const int*   l_typ   = (const int*)d_in[1];
  const float* p_x     = (const float*)d_in[2];
  const int*   p_typ   = (const int*)d_in[3];
  const float* p_score = (const float*)d_in[4];
  const float* p_rad   = (const float*)d_in[5];
  const unsigned char* l_mask = (const unsigned char*)d_in[6];
  const unsigned char* p_mask = (const unsigned char*)d_in[7];
  const float* lt_emb   = (const float*)d_in[8];
  const float* lx_w1    = (const float*)d_in[9];
  const float* lx_b1    = (const float*)d_in[10];
  const float* lx_w2    = (const float*)d_in[11];
  const float* lx_b2    = (const float*)d_in[12];
  const float* l_ln_g   = (const float*)d_in[13];
  const float* l_ln_b   = (const float*)d_in[14];
  const float* pt_emb   = (const float*)d_in[15];
  const float* px_w1    = (const float*)d_in[16];
  const float* px_b1    = (const float*)d_in[17];
  const float* px_w2    = (const float*)d_in[18];
  const float* px_b2    = (const float*)d_in[19];
  const float* ps_w1    = (const float*)d_in[20];
  const float* ps_b1    = (const float*)d_in[21];
  const float* ps_w2    = (const float*)d_in[22];
  const float* ps_b2    = (const float*)d_in[23];
  const float* p_ln_g   = (const float*)d_in[24];
  const float* p_ln_b   = (const float*)d_in[25];
  const float* proj_l_w = (const float*)d_in[26];
  const float* proj_p_w = (const float*)d_in[27];
  const float* dustbin  = (const float*)d_in[28];
  const float* sg_w1    = (const float*)d_in[29];
  const float* sg_b1    = (const float*)d_in[30];
  const float* sg_w2    = (const float*)d_in[31];
  const float* sg_b2    = (const float*)d_in[32];
  const float* retr_l_w = (const float*)d_in[33];
  const float* retr_p_w = (const float*)d_in[34];

  float* ws = (float*)d_ws;
  float* hL = ws;                                   // B*L*D   = 2,097,152
  float* hP = ws + (size_t)B_ * L_ * D_;            // B*P*D   = 16,777,216
  float* partials = hP + (size_t)B_ * P_ * D_;      // 1024*2

  float* out = (float*)d_out;
  float* o_logW   = out;                                    // B*L*(P+1)
  float* o_sharp  = out + (size_t)B_ * L_ * (P_ + 1);       // B
  float* o_negent = o_sharp + B_;                           // B
  float* o_lz     = o_negent + B_;                          // B*D
  float* o_pz     = o_lz + (size_t)B_ * D_;                 // B*D
  float* o_sig    = o_pz + (size_t)B_ * D_;                 // B*P

  encode_l_kernel<<<(B_ * L_) / 16, 256, 0, stream>>>(
      l_x, l_typ, lt_emb, lx_w1, lx_b1, lx_w2, lx_b2, l_ln_g, l_ln_b,
      proj_l_w, hL);

  encode_p_kernel<<<(B_ * P_) / 16, 256, 0, stream>>>(
      p_x, p_typ, p_score, p_rad, pt_emb, px_w1, px_b1, px_w2, px_b2,
      ps_w1, ps_b1, ps_w2, ps_b2, p_ln_g, p_ln_b, proj_p_w,
      sg_w1, sg_b1, sg_w2, sg_b2, hP, o_sig);

  logits_kernel<<<B_ * (L_ / 16), 256, 0, stream>>>(
      hL, hP, dustbin, l_mask, p_mask, o_logW, partials);

  pooled_kernel<<<2 * B_, 128, 0, stream>>>(
      hL, hP, l_mask, p_mask, retr_l_w, retr_p_w, o_lz, o_pz);

  finalize_kernel<<<1, 128, 0, stream>>>(l_mask, partials, o_sharp, o_negent);
}